// Caps_47742856462336
// MI455X (gfx1250) — compile-verified
//
#include <hip/hip_runtime.h>
#include <math.h>

typedef float v2f __attribute__((ext_vector_type(2)));
typedef float v8f __attribute__((ext_vector_type(8)));

#define BATCH    32
#define IN_CAPS  1152
#define IN_DIM   16
#define OUT_CAPS 32
#define OUT_DIM  32
#define KCHUNK   32
#define NCHUNKS  (IN_CAPS / KCHUNK)   // 36
#define LDS_PITCH 20                  // 80B rows: 16B aligned, conflict-free column reads

// ---------------------------------------------------------------------------
// CDNA5 async DMA: global -> LDS, tracked by ASYNCcnt (no VGPR round-trip).
// LDS operand = low 32 bits of the flat pointer (wave-relative LDS offset).
// ---------------------------------------------------------------------------
__device__ __forceinline__ void async_b128_to_lds(const void* gsrc, void* ldst) {
    const unsigned loff = (unsigned)(uintptr_t)ldst;
    asm volatile("global_load_async_to_lds_b128 %0, %1, off"
                 :: "v"(loff), "v"(gsrc)
                 : "memory");
}
__device__ __forceinline__ void wait_async0() {
    asm volatile("s_wait_asynccnt 0x0" ::: "memory");
}

// ---------------------------------------------------------------------------
// Fragment layout for V_WMMA_F32_16X16X4_F32 (wave32):
//   A (16x4):  lane L holds M = L%16,  K = 2*(L/16)+v   (v = vgpr 0..1)
//   B (4x16):  lane L holds N = L%16,  K = 2*(L/16)+v
//   C (16x16): vgpr r, lane L -> M = r + 8*(L/16), N = L%16
// ---------------------------------------------------------------------------

// Phase 1: S0[b,j,o] = sum_k sum_i u[b,k,i] * W[j,k,o,i]   (k-accumulated in C regs)
__global__ __launch_bounds__(128)
void caps_s0_kernel(const float* __restrict__ u, const float* __restrict__ W,
                    float* __restrict__ s0) {
    __shared__ __align__(16) float lds_u[2][32][LDS_PITCH];
    __shared__ __align__(16) float lds_w[2][32][LDS_PITCH];

    const int j    = blockIdx.y;
    const int k0   = blockIdx.x * KCHUNK;
    const int t    = threadIdx.x;
    const int wave = t >> 5;
    const int lane = t & 31;
    const int half = lane >> 4;
    const int l16  = lane & 15;
    const int b0   = (wave >> 1) * 16;
    const int n0   = (wave & 1) * 16;

    const int lb = t >> 2;          // cooperative-load row 0..31
    const int li = (t & 3) * 4;     // cooperative-load col 0,4,8,12

    // prologue: start DMA of tile 0
    async_b128_to_lds(u + ((size_t)lb * IN_CAPS + k0) * IN_DIM + li, &lds_u[0][lb][li]);
    async_b128_to_lds(W + (((size_t)j * IN_CAPS + k0) * OUT_DIM + lb) * IN_DIM + li,
                      &lds_w[0][lb][li]);

    v8f acc = {};
    for (int kk = 0; kk < KCHUNK; ++kk) {
        const int cur = kk & 1;
        wait_async0();
        __syncthreads();                 // tile[cur] ready; tile[1-cur] reads retired
        if (kk + 1 < KCHUNK) {           // kick DMA for the next tile
            const int k = k0 + kk + 1;
            async_b128_to_lds(u + ((size_t)lb * IN_CAPS + k) * IN_DIM + li,
                              &lds_u[1 - cur][lb][li]);
            async_b128_to_lds(W + (((size_t)j * IN_CAPS + k) * OUT_DIM + lb) * IN_DIM + li,
                              &lds_w[1 - cur][lb][li]);
        }
#pragma unroll
        for (int s = 0; s < 4; ++s) {
            const int ki = s * 4 + half * 2;
            v2f a, b;
            a.x = lds_u[cur][b0 + l16][ki];  a.y = lds_u[cur][b0 + l16][ki + 1];
            b.x = lds_w[cur][n0 + l16][ki];  b.y = lds_w[cur][n0 + l16][ki + 1];
            acc = __builtin_amdgcn_wmma_f32_16x16x4_f32(false, a, false, b,
                                                        (short)0, acc, false, false);
        }
    }
#pragma unroll
    for (int r = 0; r < 8; ++r) {
        const int m = b0 + r + 8 * half;
        const int o = n0 + l16;
        atomicAdd(&s0[((size_t)m * OUT_CAPS + j) * OUT_DIM + o], acc[r]);
    }
}

// Phase 2/6: squash. One wave per (b,j) row; lane = o.
__global__ __launch_bounds__(128)
void caps_squash_kernel(const float* __restrict__ s_in, const float* __restrict__ bias,
                        float* __restrict__ v_out, float scale) {
    const int idx  = blockIdx.x * 4 + (threadIdx.x >> 5);   // b*32 + j
    const int o    = threadIdx.x & 31;
    const int jj   = idx & (OUT_CAPS - 1);
    float s = s_in[(size_t)idx * OUT_DIM + o] * scale + bias[jj * OUT_DIM + o];
    float sq = s * s;
    sq += __shfl_xor(sq, 16, 32);
    sq += __shfl_xor(sq, 8, 32);
    sq += __shfl_xor(sq, 4, 32);
    sq += __shfl_xor(sq, 2, 32);
    sq += __shfl_xor(sq, 1, 32);
    const float v = (sq / (1.0f + sq)) * s * rsqrtf(sq + 1e-7f);
    v_out[(size_t)idx * OUT_DIM + o] = v;
}

// Phase 3: agree[b,k,j] = sum_o v0[b,j,o] * (W[j,k] @ u[b,k])[o]
__global__ __launch_bounds__(128)
void caps_agree_kernel(const float* __restrict__ u, const float* __restrict__ W,
                       const float* __restrict__ v0, float* __restrict__ agree) {
    __shared__ __align__(16) float lds_u[2][32][LDS_PITCH];
    __shared__ __align__(16) float lds_w[2][32][LDS_PITCH];
    __shared__ float lds_v[32][32];

    const int j    = blockIdx.y;
    const int k0   = blockIdx.x * KCHUNK;
    const int t    = threadIdx.x;
    const int wave = t >> 5;
    const int lane = t & 31;
    const int half = lane >> 4;
    const int l16  = lane & 15;
    const int b0   = (wave >> 1) * 16;
    const int n0   = (wave & 1) * 16;

    const int lb = t >> 2;
    const int li = (t & 3) * 4;

    async_b128_to_lds(u + ((size_t)lb * IN_CAPS + k0) * IN_DIM + li, &lds_u[0][lb][li]);
    async_b128_to_lds(W + (((size_t)j * IN_CAPS + k0) * OUT_DIM + lb) * IN_DIM + li,
                      &lds_w[0][lb][li]);

    // stage v0[:, j, :] (4KB) once per block
    for (int e = t; e < 1024; e += 128) {
        const int b = e >> 5, o = e & 31;
        lds_v[b][o] = v0[((size_t)b * OUT_CAPS + j) * OUT_DIM + o];
    }

    for (int kk = 0; kk < KCHUNK; ++kk) {
        const int cur = kk & 1;
        const int k   = k0 + kk;
        wait_async0();
        __syncthreads();
        if (kk + 1 < KCHUNK) {
            const int kn = k + 1;
            async_b128_to_lds(u + ((size_t)lb * IN_CAPS + kn) * IN_DIM + li,
                              &lds_u[1 - cur][lb][li]);
            async_b128_to_lds(W + (((size_t)j * IN_CAPS + kn) * OUT_DIM + lb) * IN_DIM + li,
                              &lds_w[1 - cur][lb][li]);
        }
        v8f acc = {};
#pragma unroll
        for (int s = 0; s < 4; ++s) {
            const int ki = s * 4 + half * 2;
            v2f a, b;
            a.x = lds_u[cur][b0 + l16][ki];  a.y = lds_u[cur][b0 + l16][ki + 1];
            b.x = lds_w[cur][n0 + l16][ki];  b.y = lds_w[cur][n0 + l16][ki + 1];
            acc = __builtin_amdgcn_wmma_f32_16x16x4_f32(false, a, false, b,
                                                        (short)0, acc, false, false);
        }
#pragma unroll
        for (int r = 0; r < 8; ++r) {
            const int m = b0 + r + 8 * half;
            float v = acc[r] * lds_v[m][n0 + l16];
            v += __shfl_xor(v, 8, 32);   // reduce across the 16 lanes of each half
            v += __shfl_xor(v, 4, 32);
            v += __shfl_xor(v, 2, 32);
            v += __shfl_xor(v, 1, 32);
            if (l16 == r)                // 2 lanes (one per half) commit their rows
                atomicAdd(&agree[((size_t)m * IN_CAPS + k) * OUT_CAPS + j], v);
        }
    }
}

// Phase 4: softmax over j for each (b,k); in place agree -> c. One wave per row.
__global__ __launch_bounds__(128)
void caps_softmax_kernel(float* __restrict__ a) {
    const int row  = blockIdx.x * 4 + (threadIdx.x >> 5);   // b*IN_CAPS + k
    const int lane = threadIdx.x & 31;
    float x = a[(size_t)row * OUT_CAPS + lane];
    float mx = x;
    mx = fmaxf(mx, __shfl_xor(mx, 16, 32));
    mx = fmaxf(mx, __shfl_xor(mx, 8, 32));
    mx = fmaxf(mx, __shfl_xor(mx, 4, 32));
    mx = fmaxf(mx, __shfl_xor(mx, 2, 32));
    mx = fmaxf(mx, __shfl_xor(mx, 1, 32));
    float e = __expf(x - mx);
    float sm = e;
    sm += __shfl_xor(sm, 16, 32);
    sm += __shfl_xor(sm, 8, 32);
    sm += __shfl_xor(sm, 4, 32);
    sm += __shfl_xor(sm, 2, 32);
    sm += __shfl_xor(sm, 1, 32);
    a[(size_t)row * OUT_CAPS + lane] = e / sm;
}

// Phase 5: S1[b,j,o] = sum_k c[b,k,j] * (W[j,k] @ u[b,k])[o]  (c folded into A rows)
__global__ __launch_bounds__(128)
void caps_s1_kernel(const float* __restrict__ u, const float* __restrict__ W,
                    const float* __restrict__ c, float* __restrict__ s1) {
    __shared__ __align__(16) float lds_u[2][32][LDS_PITCH];
    __shared__ __align__(16) float lds_w[2][32][LDS_PITCH];
    __shared__ float lds_c[32][33];     // [b][kk], pad 33 -> conflict-free column reads

    const int j    = blockIdx.y;
    const int k0   = blockIdx.x * KCHUNK;
    const int t    = threadIdx.x;
    const int wave = t >> 5;
    const int lane = t & 31;
    const int half = lane >> 4;
    const int l16  = lane & 15;
    const int b0   = (wave >> 1) * 16;
    const int n0   = (wave & 1) * 16;

    const int lb = t >> 2;
    const int li = (t & 3) * 4;

    async_b128_to_lds(u + ((size_t)lb * IN_CAPS + k0) * IN_DIM + li, &lds_u[0][lb][li]);
    async_b128_to_lds(W + (((size_t)j * IN_CAPS + k0) * OUT_DIM + lb) * IN_DIM + li,
                      &lds_w[0][lb][li]);

    // stage routing coefficients c[b, k0+kk, j] for the whole chunk (once per block)
    for (int e = t; e < 1024; e += 128) {
        const int b = e >> 5, kk = e & 31;
        lds_c[b][kk] = c[((size_t)b * IN_CAPS + (k0 + kk)) * OUT_CAPS + j];
    }

    v8f acc = {};
    for (int kk = 0; kk < KCHUNK; ++kk) {
        const int cur = kk & 1;
        wait_async0();
        __syncthreads();
        if (kk + 1 < KCHUNK) {
            const int k = k0 + kk + 1;
            async_b128_to_lds(u + ((size_t)lb * IN_CAPS + k) * IN_DIM + li,
                              &lds_u[1 - cur][lb][li]);
            async_b128_to_lds(W + (((size_t)j * IN_CAPS + k) * OUT_DIM + lb) * IN_DIM + li,
                              &lds_w[1 - cur][lb][li]);
        }
        // per-lane routing coefficient: A lane owns row b = b0 + l16 (both halves)
        const float cs = lds_c[b0 + l16][kk];
#pragma unroll
        for (int s = 0; s < 4; ++s) {
            const int ki = s * 4 + half * 2;
            v2f a, b;
            a.x = lds_u[cur][b0 + l16][ki] * cs;  a.y = lds_u[cur][b0 + l16][ki + 1] * cs;
            b.x = lds_w[cur][n0 + l16][ki];       b.y = lds_w[cur][n0 + l16][ki + 1];
            acc = __builtin_amdgcn_wmma_f32_16x16x4_f32(false, a, false, b,
                                                        (short)0, acc, false, false);
        }
    }
#pragma unroll
    for (int r = 0; r < 8; ++r) {
        const int m = b0 + r + 8 * half;
        const int o = n0 + l16;
        atomicAdd(&s1[((size_t)m * OUT_CAPS + j) * OUT_DIM + o], acc[r]);
    }
}

// ---------------------------------------------------------------------------
extern "C" void kernel_launch(void* const* d_in, const int* in_sizes, int n_in,
                              void* d_out, int out_size, void* d_ws, size_t ws_size,
                              hipStream_t stream) {
    const float* u    = (const float*)d_in[0];   // [32,1152,16]
    const float* W    = (const float*)d_in[1];   // [1,32,1152,32,16]
    const float* bias = (const float*)d_in[2];   // [1,32,32]
    float* out = (float*)d_out;                  // [32,32,32]

    const size_t N_S   = (size_t)BATCH * OUT_CAPS * OUT_DIM;        // 32768
    const size_t N_AGR = (size_t)BATCH * IN_CAPS * OUT_CAPS;        // 1179648
    float* s0    = (float*)d_ws;
    float* v0    = s0 + N_S;
    float* agree = v0 + N_S;
    float* s1    = agree + N_AGR;
    (void)ws_size; (void)in_sizes; (void)n_in; (void)out_size;

    hipMemsetAsync(s0,    0, N_S   * sizeof(float), stream);
    hipMemsetAsync(agree, 0, N_AGR * sizeof(float), stream);
    hipMemsetAsync(s1,    0, N_S   * sizeof(float), stream);

    const dim3 gridMM(NCHUNKS, OUT_CAPS);     // 36 x 32
    const dim3 blk(128);

    // iter 0: c == 1/32 exactly -> s0, v0
    caps_s0_kernel<<<gridMM, blk, 0, stream>>>(u, W, s0);
    caps_squash_kernel<<<dim3((BATCH * OUT_CAPS) / 4), blk, 0, stream>>>(
        s0, bias, v0, 1.0f / 32.0f);
    // agreement logits
    caps_agree_kernel<<<gridMM, blk, 0, stream>>>(u, W, v0, agree);
    // softmax over out-capsules (rows are [b,k] of 32 j's)
    caps_softmax_kernel<<<dim3((BATCH * IN_CAPS) / 4), blk, 0, stream>>>(agree);
    // iter 1: weighted sum with routed coefficients, then final squash -> output
    caps_s1_kernel<<<gridMM, blk, 0, stream>>>(u, W, agree, s1);
    caps_squash_kernel<<<dim3((BATCH * OUT_CAPS) / 4), blk, 0, stream>>>(
        s1, bias, out, 1.0f);
}